// Attention_72447508349519
// MI455X (gfx1250) — compile-verified
//
#include <hip/hip_runtime.h>
#include <hip/hip_bf16.h>

// ---------------------------------------------------------------------------
// Collapsed attention pipeline for MI455X (gfx1250, wave32).
//
// Reference einsum 'bqk,bvd->bqd' factorizes: out[b,q,d] =
// (sum_k softmax(..)[b,q,k]) * (sum_n v[b,n,d]) = 1 * colsum_v[b,d].
// So the result is q-independent and q/k/attn never need to be computed.
// Remaining work: GroupNorm row-sums -> two 16x1024x1024 GEMMs (WMMA f32)
// -> 64MB broadcast store.  HBM-bound: ~128 MB total traffic.
// ---------------------------------------------------------------------------

typedef __attribute__((ext_vector_type(2))) float v2f;
typedef __attribute__((ext_vector_type(8))) float v8f;

#define B_SZ   16
#define DIMN   1024
#define NGRP   32
#define GWID   32          // n-columns per group (1024/32)

// --- Kernel 1: per-(b,g) GroupNorm statistics over 32*1024 = 32768 elems ---
// x layout: (b, c, n) with n = h*w; group g covers n in [32g, 32g+32), all c.
__global__ void gn_stats_kernel(const float* __restrict__ x,
                                float* __restrict__ mu,
                                float* __restrict__ rs) {
    const int b = blockIdx.x, g = blockIdx.y;
    const float* xb = x + (size_t)b * DIMN * DIMN;
    const int t = threadIdx.x;           // 256 threads
    const int nloc = t & 31;             // n within group band
    const int crow = t >> 5;             // 8 c-rows per sweep
    float s = 0.f, s2 = 0.f;
    for (int c = crow; c < DIMN; c += 8) {
        float v = xb[(size_t)c * DIMN + g * GWID + nloc];
        s += v; s2 += v * v;
    }
    __shared__ float ls[256], ls2[256];
    ls[t] = s; ls2[t] = s2;
    __syncthreads();
    for (int off = 128; off > 0; off >>= 1) {
        if (t < off) { ls[t] += ls[t + off]; ls2[t] += ls2[t + off]; }
        __syncthreads();
    }
    if (t == 0) {
        const float inv = 1.0f / 32768.0f;
        float mean = ls[0] * inv;
        float var  = ls2[0] * inv - mean * mean;
        mu[b * NGRP + g] = mean;
        rs[b * NGRP + g] = rsqrtf(var + 1e-5f);
    }
}

// --- Kernel 2: srow[b,c] = sum_n norm_x[b,n,c]  (one wave32 per (b,c) row) --
// = sum_g rs[b,g]*(sum_{n in g} gamma[n]*x[b,c,n] - mu[b,g]*sum gamma) + sum beta
__global__ void row_sum_kernel(const float* __restrict__ x,
                               const float* __restrict__ gamma,
                               const float* __restrict__ beta,
                               const float* __restrict__ mu,
                               const float* __restrict__ rs,
                               float* __restrict__ srow) {
    const int row  = blockIdx.x * (blockDim.x >> 5) + (threadIdx.x >> 5);
    const int lane = threadIdx.x & 31;   // lane == group id
    const int b = row >> 10, c = row & 1023;
    const float* xr = x + (size_t)b * DIMN * DIMN + (size_t)c * DIMN + lane * GWID;
    const float* gm = gamma + lane * GWID;
    const float* bt = beta  + lane * GWID;
    float wsum = 0.f, gsum = 0.f, bsum = 0.f;
#pragma unroll 8
    for (int j = 0; j < GWID; ++j) {
        float g = gm[j];
        wsum += g * xr[j];
        gsum += g;
        bsum += bt[j];
    }
    float val = rs[b * NGRP + lane] * (wsum - mu[b * NGRP + lane] * gsum) + bsum;
    // wave32 butterfly reduce
#pragma unroll
    for (int off = 16; off > 0; off >>= 1)
        val += __shfl_xor(val, off, 32);
    if (lane == 0) srow[row] = val;
}

// --- Kernels 3/4: D(16x1024) = A(16x1024) @ W(1024x1024)^T + scale*bias ----
// Full-fp32 WMMA: V_WMMA_F32_16X16X4_F32, one 16x16 tile per wave,
// 256 wmma issues over K=1024.  64 waves total (16 blocks x 4 waves).
// ISA 16x4 f32 A layout: lanes 0-15 hold (M=lane, K=0,1) in v[0],v[1];
// lanes 16-31 hold K=2,3.  B(4x16) mirrored with N=lane&15.
// f32 C/D 16x16: vgpr r -> M = r + 8*(lane>=16), N = lane&15.
__global__ void gemm16_wmma_f32(const float* __restrict__ A,
                                const float* __restrict__ W,
                                const float* __restrict__ bias,
                                float biasScale,
                                float* __restrict__ Out) {
    const int lane  = threadIdx.x & 31;
    const int wid   = (blockIdx.x * blockDim.x + threadIdx.x) >> 5;  // 0..63
    const int m     = lane & 15;
    const int n     = wid * 16 + (lane & 15);
    const int khalf = (lane >> 4) * 2;   // 0 for lanes 0-15, 2 for 16-31
    const float* Arow = A + (size_t)m * DIMN + khalf;
    const float* Wrow = W + (size_t)n * DIMN + khalf;

    v8f acc = {};
    for (int k0 = 0; k0 < DIMN; k0 += 4) {
        __builtin_prefetch(&Wrow[k0 + 128], 0, 0);   // global_prefetch_b8
        v2f a, bfrag;
        a[0]     = Arow[k0 + 0];
        a[1]     = Arow[k0 + 1];
        bfrag[0] = Wrow[k0 + 0];
        bfrag[1] = Wrow[k0 + 1];
        acc = __builtin_amdgcn_wmma_f32_16x16x4_f32(
                  /*neg_a=*/false, a, /*neg_b=*/false, bfrag,
                  /*c_mod=*/(short)0, acc, /*reuse_a=*/false, /*reuse_b=*/false);
    }
    const float bn = bias[n] * biasScale;
    const int rbase = (lane >> 4) * 8;
#pragma unroll
    for (int r = 0; r < 8; ++r)
        Out[(size_t)(rbase + r) * DIMN + n] = acc[r] + bn;
}

// --- Kernel 5: d_out[(b*1024 + d)*1024 + n] = y[b*1024 + d]  (64 MB, b128) --
__global__ void broadcast_out_kernel(const float* __restrict__ y,
                                     float* __restrict__ out) {
    const int row = blockIdx.x;                 // b*1024 + d, 0..16383
    const float v = y[row];
    float4 v4 = make_float4(v, v, v, v);
    float4* o = (float4*)(out + (size_t)row * DIMN);
    o[threadIdx.x] = v4;                        // 256 threads x 16B = 4 KB row
}

extern "C" void kernel_launch(void* const* d_in, const int* in_sizes, int n_in,
                              void* d_out, int out_size, void* d_ws, size_t ws_size,
                              hipStream_t stream) {
    // setup_inputs order: x, gamma, beta, Wq, bq, Wk, bk, Wv, bv, Wo, bo
    const float* x     = (const float*)d_in[0];
    const float* gamma = (const float*)d_in[1];
    const float* beta  = (const float*)d_in[2];
    // Wq/bq (3,4) and Wk/bk (5,6) are mathematically dead: softmax rowsum == 1.
    const float* Wv    = (const float*)d_in[7];
    const float* bv    = (const float*)d_in[8];
    const float* Wo    = (const float*)d_in[9];
    const float* bo    = (const float*)d_in[10];
    float* out = (float*)d_out;

    float* mu   = (float*)d_ws;          // 16*32
    float* rs   = mu   + B_SZ * NGRP;    // 16*32
    float* srow = rs   + B_SZ * NGRP;    // 16*1024
    float* cv   = srow + B_SZ * DIMN;    // 16*1024  (colsum of v)
    float* y    = cv   + B_SZ * DIMN;    // 16*1024  (final row values)

    gn_stats_kernel<<<dim3(B_SZ, NGRP), 256, 0, stream>>>(x, mu, rs);
    row_sum_kernel<<<(B_SZ * DIMN) / 8, 256, 0, stream>>>(x, gamma, beta, mu, rs, srow);
    // cv[b,d] = srow[b,:] . Wv[d,:] + 1024*bv[d]
    gemm16_wmma_f32<<<16, 128, 0, stream>>>(srow, Wv, bv, (float)DIMN, cv);
    // y[b,e] = cv[b,:] . Wo[e,:] + bo[e]
    gemm16_wmma_f32<<<16, 128, 0, stream>>>(cv, Wo, bo, 1.0f, y);
    broadcast_out_kernel<<<B_SZ * DIMN, 256, 0, stream>>>(y, out);
}